// FragmentsizeDistribution5_64802466562906
// MI455X (gfx1250) — compile-verified
//
#include <hip/hip_runtime.h>

typedef float v2f __attribute__((ext_vector_type(2)));
typedef float v8f __attribute__((ext_vector_type(8)));
typedef int   v2i __attribute__((ext_vector_type(2)));

#define IN_DIM  10
#define OUT_DIM 8
#define TILE_ELEMS (IN_DIM * OUT_DIM)   // 80 floats per (head|head,split) tile
#define TPG 4                            // tokens per WMMA group (one 16x16x4 chain)
#define NG  2                            // groups per wave -> 8 tokens per wave

// One wave32 computes NG*TPG = 8 tokens via NG independent chains of
// 10 V_WMMA_F32_16X16X4_F32 each:
//   A (16x4): diagonal — A[m,m] = x[t+m, j] for m=0..3 (lanes 0,1,18,19 per ISA layout)
//   B (4x16): row m    — B[m,n] = W[h_m][j,n] + 0.1*dW[h_m*ns+s_m][j,n]
//   D[m,n] accumulates over j -> out[t+m, n] in acc VGPR m, lanes 0..7.
//
// B columns n>=8 carry clamped-duplicate garbage: harmless, since D column n
// depends only on B column n and columns 8..15 are never stored. A rows 0..3
// must be exact, hence the ax/ay masks. All loads are unconditional at
// clamped addresses; EXEC stays all-ones around every WMMA.
__global__ __launch_bounds__(256)
void frag_dist5_wmma_kernel(const float* __restrict__ x,
                            const float* __restrict__ W,
                            const float* __restrict__ dW,
                            const int*   __restrict__ head_ix,
                            const int*   __restrict__ split_ix,
                            const int*   __restrict__ n_splits_p,
                            float*       __restrict__ out,
                            int Btot)
{
    const int ns   = n_splits_p[0];
    const int lane = threadIdx.x & 31;
    const int waveInBlock   = threadIdx.x >> 5;
    const int wavesPerBlock = blockDim.x >> 5;
    const int wid    = blockIdx.x * wavesPerBlock + waveInBlock;
    const int nWaves = gridDim.x * wavesPerBlock;

    const int  n        = lane & 15;                     // B-matrix column (N)
    const int  sel      = lane >> 4;                     // 0 -> K rows {0,1}, 1 -> {2,3}
    const int  nc       = (n < OUT_DIM) ? n : (OUT_DIM - 1);  // clamped, in-bounds
    // Diagonal x lanes (f32 16x4 A layout): lane0->(M0,K0), lane1->(M1,K1),
    // lane18->(M2,K2), lane19->(M3,K3).
    const bool  isAx   = (lane == 0) || (lane == 18);    // A VGPR0 (K = 0/2)
    const bool  isAy   = (lane == 1) || (lane == 19);    // A VGPR1 (K = 1/3)
    const float axMask = isAx ? 1.0f : 0.0f;
    const float ayMask = isAy ? 1.0f : 0.0f;

    for (int t = wid * (TPG * NG); t < Btot; t += nWaves * (TPG * NG)) {
        const float* qW0[NG]; const float* qD0[NG];
        const float* qW1[NG]; const float* qD1[NG];
        const float* px[NG];
        float w0[NG][IN_DIM], d0[NG][IN_DIM], w1[NG][IN_DIM], d1[NG][IN_DIM];
        float xr[NG][IN_DIM];

        // ---- Index gathers (paired b64 loads) + base pointers ----
        #pragma unroll
        for (int g = 0; g < NG; ++g) {
            const int tg = t + g * TPG;
            // tokens (tb0, tb0+1) feed B VGPR0/VGPR1 rows (K=2*sel, 2*sel+1).
            // tb0 is even -> 8B-aligned pair load. Clamp the PAIR base for the
            // tail; wrong-token indices there are harmless (stores are masked).
            int tb0 = tg + 2 * sel;
            tb0 = (tb0 <= Btot - 2) ? tb0 : (Btot - 2);
            const v2i hp = *(const v2i*)(head_ix + tb0);
            const v2i sp = *(const v2i*)(split_ix + tb0);
            const int h0 = hp.x, h1 = hp.y, s0 = sp.x, s1 = sp.y;
            qW0[g] = W  + (size_t)h0 * TILE_ELEMS + nc;
            qD0[g] = dW + ((size_t)h0 * ns + s0) * TILE_ELEMS + nc;
            qW1[g] = W  + (size_t)h1 * TILE_ELEMS + nc;
            qD1[g] = dW + ((size_t)h1 * ns + s1) * TILE_ELEMS + nc;
            // Ay-lanes read tb1's x row, everyone else tb0's (broadcast, coalesces).
            px[g]  = x + (size_t)(isAy ? (tb0 + 1) : tb0) * IN_DIM;
        }

        // ---- Issue ALL gathers up front (max loads in flight) ----
        #pragma unroll
        for (int g = 0; g < NG; ++g) {
            #pragma unroll
            for (int j = 0; j < IN_DIM; ++j) {
                w0[g][j] = qW0[g][j * OUT_DIM];
                d0[g][j] = qD0[g][j * OUT_DIM];
                w1[g][j] = qW1[g][j * OUT_DIM];
                d1[g][j] = qD1[g][j * OUT_DIM];
            }
            #pragma unroll
            for (int j = 0; j < IN_DIM; ++j) xr[g][j] = px[g][j];   // vectorizes to b128
        }

        // ---- Fuse w = W + 0.1*dW and run two independent WMMA chains ----
        v8f acc[NG];
        #pragma unroll
        for (int g = 0; g < NG; ++g) acc[g] = (v8f){};
        #pragma unroll
        for (int j = 0; j < IN_DIM; ++j) {
            #pragma unroll
            for (int g = 0; g < NG; ++g) {
                v2f a, b;
                a.x = xr[g][j] * axMask;               // K = 0/2 diagonal slot
                a.y = xr[g][j] * ayMask;               // K = 1/3 diagonal slot
                b.x = fmaf(0.1f, d0[g][j], w0[g][j]);  // B row K = 2*sel (cols>=8 don't-care)
                b.y = fmaf(0.1f, d1[g][j], w1[g][j]);  // B row K = 2*sel+1
                acc[g] = __builtin_amdgcn_wmma_f32_16x16x4_f32(
                             /*neg_a=*/false, a, /*neg_b=*/false, b,
                             /*c_mod=*/(short)0, acc[g], /*reuse_a=*/false, /*reuse_b=*/false);
            }
        }

        // D layout: VGPR m holds M=m on lanes 0-15 (N=lane).
        // Token t+g*TPG+m -> acc[g][m], lanes 0..7.
        if (lane < OUT_DIM) {
            #pragma unroll
            for (int g = 0; g < NG; ++g) {
                #pragma unroll
                for (int m = 0; m < TPG; ++m) {
                    const int tok = t + g * TPG + m;
                    if (tok < Btot)
                        out[(size_t)tok * OUT_DIM + lane] = acc[g][m];
                }
            }
        }
    }
}

extern "C" void kernel_launch(void* const* d_in, const int* in_sizes, int n_in,
                              void* d_out, int out_size, void* d_ws, size_t ws_size,
                              hipStream_t stream) {
    const float* x   = (const float*)d_in[0];
    const float* W   = (const float*)d_in[1];
    const float* dW  = (const float*)d_in[2];
    const int*   hix = (const int*)d_in[3];
    const int*   six = (const int*)d_in[4];
    const int*   nsp = (const int*)d_in[5];
    float* out = (float*)d_out;

    const int Btot = in_sizes[0] / IN_DIM;                    // 2,000,000
    const int tokensPerWave = TPG * NG;                       // 8
    const int waveTiles = (Btot + tokensPerWave - 1) / tokensPerWave;
    const int wavesPerBlock = 8;                              // 256 threads = 8 wave32
    const int blocks = (waveTiles + wavesPerBlock - 1) / wavesPerBlock;

    frag_dist5_wmma_kernel<<<blocks, 256, 0, stream>>>(x, W, dW, hix, six, nsp, out, Btot);
}